// TaskAlignedAssigner_68788196212955
// MI455X (gfx1250) — compile-verified
//
#include <hip/hip_runtime.h>
#include <hip/hip_bf16.h>

#define B_ 16
#define N_ 8400
#define M_ 32
#define C_ 80
#define TOPK_ 13
#define EPS_ 1e-9f

typedef __attribute__((ext_vector_type(2))) float v2f;
typedef __attribute__((ext_vector_type(8))) float v8f;

// CIoU clipped to >= 0, exactly mirroring the reference math (eps = 1e-7).
__device__ __forceinline__ float ciou_clip(float gx1, float gy1, float gx2, float gy2,
                                           float px1, float py1, float px2, float py2) {
  const float eps = 1e-7f;
  float iw = fmaxf(fminf(gx2, px2) - fmaxf(gx1, px1), 0.0f);
  float ih = fmaxf(fminf(gy2, py2) - fmaxf(gy1, py1), 0.0f);
  float inter = iw * ih;
  float w1 = gx2 - gx1, h1 = gy2 - gy1 + eps;
  float w2 = px2 - px1, h2 = py2 - py1 + eps;
  float uni = w1 * h1 + w2 * h2 - inter + eps;
  float iou = inter / uni;
  float cw = fmaxf(gx2, px2) - fminf(gx1, px1);
  float ch = fmaxf(gy2, py2) - fminf(gy1, py1);
  float c2 = cw * cw + ch * ch + eps;
  float dx = px1 + px2 - gx1 - gx2;
  float dy = py1 + py2 - gy1 - gy2;
  float rho2 = (dx * dx + dy * dy) * 0.25f;
  float d = atanf(w2 / h2) - atanf(w1 / h1);
  const float kpi = 0.40528473456935109f; // 4 / pi^2
  float v = kpi * d * d;
  float a = v / (v - iou + 1.0f + eps);
  float ci = iou - (rho2 / c2 + v * a);
  return fmaxf(ci, 0.0f);
}

__device__ __forceinline__ float ingt_dmin(float ax, float ay,
                                           float gx1, float gy1, float gx2, float gy2) {
  return fminf(fminf(ax - gx1, ay - gy1), fminf(gx2 - ax, gy2 - ay));
}

// ---------------------------------------------------------------------------
// K0: zero the accumulating workspace (mask bits + per-row maxima).
// ---------------------------------------------------------------------------
__global__ void k_init(unsigned* __restrict__ bits, unsigned* __restrict__ posA,
                       unsigned* __restrict__ posO) {
  int i = blockIdx.x * blockDim.x + threadIdx.x;
  if (i < B_ * N_) bits[i] = 0u;
  if (i < B_ * M_) { posA[i] = 0u; posO[i] = 0u; }
}

// ---------------------------------------------------------------------------
// K1: bbox_scores gather as a one-hot GEMM on the WMMA pipe.
//     G[b,m,n] = pd_scores[b,n,cls[b,m]]  ==  onehot(cls) (M x 80) @ scores^T.
//     Exact in fp32 (products are 0 or the score itself).
//     Block = 64 threads = 2 full wave32s, one 16-row m-tile per wave.
// ---------------------------------------------------------------------------
__global__ void __launch_bounds__(64) k_gather_wmma(const float* __restrict__ scores,
                                                    const int* __restrict__ labels,
                                                    float* __restrict__ G) {
  const int nTiles = N_ / 16;        // 525
  int b  = blockIdx.x / nTiles;
  int nt = blockIdx.x % nTiles;

  __shared__ float S[16 * 84];       // 16 anchors x 80 classes, pitch 84 (conflict-free)

  const float* src = scores + ((size_t)b * N_ + (size_t)nt * 16) * C_; // contiguous 1280 floats
  for (int i = threadIdx.x; i < 16 * C_; i += 64) {
    int n = i / C_, c = i % C_;
    S[n * 84 + c] = src[i];
  }
  __syncthreads();

  int lane = threadIdx.x & 31;
  int wv   = threadIdx.x >> 5;       // which m-tile (0/1)
  int mrow = wv * 16 + (lane & 15);
  int cls  = labels[b * M_ + mrow];
  int kh   = lane >> 4;              // K half select per ISA A/B layout
  int ncol = lane & 15;

  v8f acc = {};
  #pragma unroll
  for (int s = 0; s < C_ / 4; ++s) {
    int k0 = 4 * s + 2 * kh;
    v2f a, bb;
    a.x  = (cls == k0)     ? 1.0f : 0.0f;
    a.y  = (cls == k0 + 1) ? 1.0f : 0.0f;
    bb.x = S[ncol * 84 + k0];
    bb.y = S[ncol * 84 + k0 + 1];
    acc = __builtin_amdgcn_wmma_f32_16x16x4_f32(false, a, false, bb,
                                                (short)0, acc, false, false);
  }

  int nout = nt * 16 + ncol;
  #pragma unroll
  for (int r = 0; r < 8; ++r) {
    int mo = wv * 16 + r + 8 * kh;   // C/D layout: VGPR r, lanes 16-31 -> M+8
    G[((size_t)b * M_ + mo) * N_ + nout] = acc[r];
  }
}

// ---------------------------------------------------------------------------
// K2: per (b,m) row: align_metric into LDS (33.6KB of the 320KB WGP LDS),
//     then 13 iterative argmax reductions (jax.lax.top_k tie order: smaller
//     index first). Winners that pass the validity test set bit m of the
//     per-anchor mask.
// ---------------------------------------------------------------------------
__global__ void __launch_bounds__(256) k_rowtopk(const float* __restrict__ pdb,
                                                 const float* __restrict__ anc,
                                                 const float* __restrict__ gtb,
                                                 const int* __restrict__ mgt,
                                                 const float* __restrict__ G,
                                                 unsigned* __restrict__ bits) {
  int row = blockIdx.x;              // b*M + m
  int b = row / M_, m = row % M_;
  if (mgt[row] == 0) return;         // uniform: invalid gt contributes nothing

  __shared__ float vals[N_];
  __shared__ unsigned long long red[256];

  float gx1 = gtb[row * 4 + 0], gy1 = gtb[row * 4 + 1];
  float gx2 = gtb[row * 4 + 2], gy2 = gtb[row * 4 + 3];
  const float* Grow = G + (size_t)row * N_;
  const float* pb   = pdb + (size_t)b * N_ * 4;

  for (int n = threadIdx.x; n < N_; n += 256) {
    __builtin_prefetch(pb + 4 * (n + 1024), 0, 1);   // global_prefetch_b8
    float ax = anc[2 * n], ay = anc[2 * n + 1];
    float metric = 0.0f;
    if (ingt_dmin(ax, ay, gx1, gy1, gx2, gy2) > 1e-9f) {
      float4 p = *(const float4*)(pb + 4 * n);
      float ov  = ciou_clip(gx1, gy1, gx2, gy2, p.x, p.y, p.z, p.w);
      float ov2 = ov * ov;
      metric = Grow[n] * (ov2 * ov2 * ov2);          // score^1 * overlap^6
    }
    vals[n] = metric;                                 // >= 0 always
  }
  __syncthreads();

  for (int it = 0; it < TOPK_; ++it) {
    float bv = -1.0f;  int bi = 0;                    // -1 marks picked slots
    for (int n = threadIdx.x; n < N_; n += 256) {
      float v = vals[n];
      if (v > bv) { bv = v; bi = n; }                 // ascending scan: first occurrence
    }
    // key: (value bits << 32) | (N - idx); values >= 0 so uint order == float order,
    // and ties prefer the smaller index.
    red[threadIdx.x] =
        ((unsigned long long)__float_as_uint(fmaxf(bv, 0.0f)) << 32) |
        (unsigned)(N_ - bi);
    __syncthreads();
    for (int s = 128; s > 0; s >>= 1) {
      if (threadIdx.x < s && red[threadIdx.x + s] > red[threadIdx.x])
        red[threadIdx.x] = red[threadIdx.x + s];
      __syncthreads();
    }
    if (threadIdx.x == 0) {
      int nw = N_ - (int)(unsigned)(red[0] & 0xffffffffu);
      vals[nw] = -1.0f;                               // never re-picked
      float ax = anc[2 * nw], ay = anc[2 * nw + 1];
      if (ingt_dmin(ax, ay, gx1, gy1, gx2, gy2) > 1e-9f)  // mask_pos = topk & valid
        atomicOr(&bits[b * N_ + nw], 1u << m);
    }
    __syncthreads();
  }
}

// ---------------------------------------------------------------------------
// K3: per (b,n): conflict resolution (fg_count>1 -> argmax overlap over all m,
//     first occurrence), target gt, masked align/overlap at the chosen gt,
//     atomicMax (float-bits, all values >= 0) into per-row pos_align/pos_over.
// ---------------------------------------------------------------------------
__global__ void __launch_bounds__(256) k_resolve(const float* __restrict__ pdb,
                                                 const float* __restrict__ anc,
                                                 const float* __restrict__ gtb,
                                                 const int* __restrict__ mgt,
                                                 const float* __restrict__ G,
                                                 const unsigned* __restrict__ bits,
                                                 unsigned* __restrict__ posA,
                                                 unsigned* __restrict__ posO,
                                                 int* __restrict__ tgtfg,
                                                 float* __restrict__ amv) {
  int g = blockIdx.x * blockDim.x + threadIdx.x;
  if (g >= B_ * N_) return;
  int b = g / N_, n = g % N_;

  unsigned bm = bits[g];
  float4 p = *(const float4*)(pdb + ((size_t)b * N_ + n) * 4);
  float ax = anc[2 * n], ay = anc[2 * n + 1];

  if (__popc(bm) > 1) {                       // rebuild column as one-hot at best overlap
    float bestv = -1.0f; int bestm = 0;
    for (int m = 0; m < M_; ++m) {
      int row = b * M_ + m;
      float gx1 = gtb[row * 4], gy1 = gtb[row * 4 + 1];
      float gx2 = gtb[row * 4 + 2], gy2 = gtb[row * 4 + 3];
      float ov = 0.0f;
      if (mgt[row] != 0 && ingt_dmin(ax, ay, gx1, gy1, gx2, gy2) > 1e-9f)
        ov = ciou_clip(gx1, gy1, gx2, gy2, p.x, p.y, p.z, p.w);
      if (ov > bestv) { bestv = ov; bestm = m; }      // jnp.argmax: first max
    }
    bm = 1u << bestm;
  }

  int fg  = (bm != 0u);
  int tgt = fg ? (__ffs(bm) - 1) : 0;
  float am = 0.0f;
  if (fg) {
    int row = b * M_ + tgt;
    float gx1 = gtb[row * 4], gy1 = gtb[row * 4 + 1];
    float gx2 = gtb[row * 4 + 2], gy2 = gtb[row * 4 + 3];
    float ov = 0.0f;
    if (mgt[row] != 0 && ingt_dmin(ax, ay, gx1, gy1, gx2, gy2) > 1e-9f) {
      ov = ciou_clip(gx1, gy1, gx2, gy2, p.x, p.y, p.z, p.w);
      float ov2 = ov * ov;
      am = G[(size_t)row * N_ + n] * (ov2 * ov2 * ov2);
    }
    atomicMax(&posA[row], __float_as_uint(am));       // values >= 0: uint order ok
    atomicMax(&posO[row], __float_as_uint(ov));
  }
  tgtfg[g] = fg ? (256 | tgt) : 0;
  amv[g]   = am;
}

// ---------------------------------------------------------------------------
// K4: final write-out. 64 anchors per block; the 80-wide one-hot score rows
//     are contiguous in memory -> coalesced non-temporal streaming stores
//     (47MB of writes is the roofline cost of this whole workload).
// ---------------------------------------------------------------------------
__global__ void __launch_bounds__(256) k_write(const int* __restrict__ labels,
                                               const float* __restrict__ gtb,
                                               const unsigned* __restrict__ posA,
                                               const unsigned* __restrict__ posO,
                                               const int* __restrict__ tgtfg,
                                               const float* __restrict__ amv,
                                               float* __restrict__ out) {
  const int TILE = 64;
  const int nTiles = (N_ + TILE - 1) / TILE;
  int b  = blockIdx.x / nTiles;
  int t  = blockIdx.x % nTiles;
  int n0 = t * TILE;

  __shared__ int   lab[TILE];
  __shared__ float nrm[TILE];

  float* oL = out;                                    // target_labels  (B,N)
  float* oB = out + (size_t)B_ * N_;                  // target_bboxes  (B,N,4)
  float* oS = oB + (size_t)B_ * N_ * 4;               // target_scores  (B,N,C)
  float* oF = oS + (size_t)B_ * N_ * C_;              // fg_mask        (B,N)
  float* oT = oF + (size_t)B_ * N_;                   // target_gt_idx  (B,N)

  if (threadIdx.x < TILE) {
    int a = n0 + threadIdx.x;
    if (a < N_) {
      int g    = b * N_ + a;
      int code = tgtfg[g];
      int fg   = code >> 8;
      int tgt  = code & 255;
      int row  = b * M_ + tgt;
      float norm = 0.0f;
      if (fg)
        norm = amv[g] * __uint_as_float(posO[row]) /
               (__uint_as_float(posA[row]) + EPS_);
      int lb = labels[row];
      lab[threadIdx.x] = lb;
      nrm[threadIdx.x] = norm;

      __builtin_nontemporal_store((float)lb, &oL[g]);
      float4 box = *(const float4*)(gtb + row * 4);
      __builtin_nontemporal_store(box.x, &oB[(size_t)g * 4 + 0]);
      __builtin_nontemporal_store(box.y, &oB[(size_t)g * 4 + 1]);
      __builtin_nontemporal_store(box.z, &oB[(size_t)g * 4 + 2]);
      __builtin_nontemporal_store(box.w, &oB[(size_t)g * 4 + 3]);
      __builtin_nontemporal_store(fg ? 1.0f : 0.0f, &oF[g]);
      __builtin_nontemporal_store((float)tgt, &oT[g]);
    }
  }
  __syncthreads();

  for (int i = threadIdx.x; i < TILE * C_; i += 256) {
    int a = i / C_, c = i % C_;
    int n = n0 + a;
    if (n < N_) {
      float v = (c == lab[a]) ? nrm[a] : 0.0f;
      __builtin_nontemporal_store(v, &oS[((size_t)(b * N_ + n)) * C_ + c]);
    }
  }
}

extern "C" void kernel_launch(void* const* d_in, const int* in_sizes, int n_in,
                              void* d_out, int out_size, void* d_ws, size_t ws_size,
                              hipStream_t stream) {
  const float* pd_scores = (const float*)d_in[0];
  const float* pd_bboxes = (const float*)d_in[1];
  const float* anc       = (const float*)d_in[2];
  const int*   gt_labels = (const int*)d_in[3];
  const float* gt_bboxes = (const float*)d_in[4];
  const int*   mask_gt   = (const int*)d_in[5];
  float* out = (float*)d_out;

  char* ws = (char*)d_ws;
  float*    G     = (float*)ws;     ws += sizeof(float)    * (size_t)B_ * M_ * N_;
  unsigned* bits  = (unsigned*)ws;  ws += sizeof(unsigned) * (size_t)B_ * N_;
  unsigned* posA  = (unsigned*)ws;  ws += sizeof(unsigned) * (size_t)B_ * M_;
  unsigned* posO  = (unsigned*)ws;  ws += sizeof(unsigned) * (size_t)B_ * M_;
  int*      tgtfg = (int*)ws;       ws += sizeof(int)      * (size_t)B_ * N_;
  float*    amv   = (float*)ws;

  hipLaunchKernelGGL(k_init, dim3((B_ * N_ + 255) / 256), dim3(256), 0, stream,
                     bits, posA, posO);
  hipLaunchKernelGGL(k_gather_wmma, dim3(B_ * (N_ / 16)), dim3(64), 0, stream,
                     pd_scores, gt_labels, G);
  hipLaunchKernelGGL(k_rowtopk, dim3(B_ * M_), dim3(256), 0, stream,
                     pd_bboxes, anc, gt_bboxes, mask_gt, G, bits);
  hipLaunchKernelGGL(k_resolve, dim3((B_ * N_ + 255) / 256), dim3(256), 0, stream,
                     pd_bboxes, anc, gt_bboxes, mask_gt, G, bits, posA, posO, tgtfg, amv);
  hipLaunchKernelGGL(k_write, dim3(B_ * ((N_ + 63) / 64)), dim3(256), 0, stream,
                     gt_labels, gt_bboxes, posA, posO, tgtfg, amv, out);
}